// GraphMomentAggregator_9801115369801
// MI455X (gfx1250) — compile-verified
//
#include <hip/hip_runtime.h>

// GraphMomentAggregator for MI455X (gfx1250, wave32).
// Roofline: 204.8 MB in / 0.5 MB out -> ~8.8 us floor at 23.3 TB/s; compute ~0.36 GFLOP.
// Segment-sum realized as one-hot matmul on V_WMMA_F32_16X16X4_F32 (full fp32).
// Data movement: Tensor Data Mover stages 24x256 f32 tiles global->LDS (double
// buffered, TENSORcnt-tracked); waves feed WMMA B-tiles from LDS via ds_load_b32.

typedef __attribute__((ext_vector_type(2))) float v2f;
typedef __attribute__((ext_vector_type(8))) float v8f;
typedef __attribute__((ext_vector_type(4))) unsigned int v4u;
typedef __attribute__((ext_vector_type(4))) int v4i;
typedef __attribute__((ext_vector_type(8))) int v8i;

#define NUM_GRAPHS 64
#define FEAT 256            // K*M*F = 4*4*16 contiguous floats per (o, node)
#define O_DIM 2
#define S_MOM 4
#define TNODES 24           // nodes per TDM tile (24*256*4B = 24 KB per buffer)
#define CHUNK (TNODES * 10) // 240 nodes per block -> all DMA tiles full except grid tail

#if __has_builtin(__builtin_amdgcn_tensor_load_to_lds)
#define HAVE_TDM 1
#else
#define HAVE_TDM 0
#endif

__global__ void __launch_bounds__(512) zero_kernel(float* out, int n) {
  int i = blockIdx.x * blockDim.x + threadIdx.x;
  if (i < n) out[i] = 0.0f;
}

#if HAVE_TDM
// Issue one TDM descriptor: 2-D tensor (dim0=256 cols, dim1=rows), 4-byte
// elements, tile = 256 x TNODES, contiguous rows (stride0 = 256). Rows past
// `remaining_rows` are OOB -> DMA writes zeros (never read by compute).
__device__ __forceinline__ void tdm_load_tile(const float* gsrc, unsigned lds_off,
                                              int remaining_rows) {
  unsigned long long ga = (unsigned long long)(uintptr_t)gsrc;
  v4u g0;
  g0.x = 1u;                                                // count=1 (valid D#)
  g0.y = lds_off;                                           // lds_addr (bytes)
  g0.z = (unsigned)ga;                                      // global_addr[31:0]
  g0.w = ((unsigned)(ga >> 32) & 0x01ffffffu) | (2u << 30); // ga[56:32] | type=2
  v8i g1;
  g1[0] = (int)(2u << 16);                  // data_size=4B; no pad/iter/multicast
  g1[1] = (int)((unsigned)FEAT << 16);      // tensor_dim0[15:0] (=256)
  g1[2] = (int)(((unsigned)remaining_rows & 0xffffu) << 16);            // td1 lo
  g1[3] = (int)(((unsigned)remaining_rows >> 16) | ((unsigned)FEAT << 16)); // td1 hi | tile_dim0
  g1[4] = TNODES;                           // tile_dim1 | tile_dim2=0
  g1[5] = FEAT;                             // tensor_dim0_stride[31:0]
  g1[6] = 0;
  g1[7] = 0;
  v4i z4 = {0, 0, 0, 0};
#if defined(__clang_major__) && __clang_major__ >= 23
  v8i z8 = {0, 0, 0, 0, 0, 0, 0, 0};
  __builtin_amdgcn_tensor_load_to_lds(g0, g1, z4, z4, z8, 0);
#else
  __builtin_amdgcn_tensor_load_to_lds(g0, g1, z4, z4, 0);
#endif
}
#endif

__global__ void __launch_bounds__(512)
moment_seg_wmma(const float* __restrict__ x,
                const int*   __restrict__ bi,
                float*       __restrict__ out,
                int n_nodes)
{
  const int lane = threadIdx.x & 31;
  const int wave = threadIdx.x >> 5;        // 16 waves: one 16-column feature group each
  const int o    = blockIdx.y;              // 0..1
  const int row  = lane & 15;               // graph row in 16-row window == col in group
  const int h    = lane >> 4;               // half-wave selector
  const int col  = wave * 16 + row;         // feature column 0..255

  const int c0 = blockIdx.x * CHUNK;
  if (c0 >= n_nodes) return;
  const int c1    = min(c0 + CHUNK, n_nodes);
  const int nodes = c1 - c0;
  const int nfull = c0 + (nodes & ~3);

  const float* xo = x + (size_t)o * (size_t)n_nodes * FEAT;

  v8f acc1 = {}, acc2 = {}, acc3 = {}, acc4 = {};
  int g_base = bi[c0];

  auto flush = [&]() {
    // C/D layout: lane L, VGPR j -> M = j + (L>=16 ? 8 : 0), N = L&15
#pragma unroll
    for (int j = 0; j < 8; ++j) {
      int g = g_base + j + (h ? 8 : 0);
      if (g < NUM_GRAPHS) {
        size_t base = (((size_t)g * O_DIM + o) * S_MOM) * FEAT + col;
        float f1 = acc1[j], f2 = acc2[j], f3 = acc3[j], f4 = acc4[j];
        if (f1 != 0.0f) atomicAdd(&out[base],            f1);
        if (f2 != 0.0f) atomicAdd(&out[base + FEAT],     f2);
        if (f3 != 0.0f) atomicAdd(&out[base + 2 * FEAT], f3);
        if (f4 != 0.0f) atomicAdd(&out[base + 3 * FEAT], f4);
      }
    }
    acc1 = {}; acc2 = {}; acc3 = {}; acc4 = {};
  };

  // One 4-node WMMA step; vA/vB are |x| for this lane's two K-slots
  // (A layout: VGPR0 <-> K=2h, VGPR1 <-> K=2h+1 ; nodes n0+2h, n0+2h+1).
  auto step = [&](int n0, float vA, float vB) {
    int4 g4 = *(const int4*)(bi + n0);                    // wave-uniform -> s_load_b128
    if (g4.w - g_base > 15) { flush(); g_base = g4.x; }   // sorted index: rare, uniform
    const int gA = h ? g4.z : g4.x;
    const int gB = h ? g4.w : g4.y;

    const float vA2 = vA * vA, vB2 = vB * vB;

    v2f a;                                                // one-hot selection matrix
    a.x = (gA == g_base + row) ? 1.0f : 0.0f;
    a.y = (gB == g_base + row) ? 1.0f : 0.0f;

    // Extremely-rare fallback: graph jumped past the 16-row window
    if (__builtin_expect((unsigned)(gA - g_base) >= 16u, 0)) {
      size_t base = (((size_t)gA * O_DIM + o) * S_MOM) * FEAT + col;
      atomicAdd(&out[base], vA);                  atomicAdd(&out[base + FEAT], vA2);
      atomicAdd(&out[base + 2 * FEAT], vA2 * vA); atomicAdd(&out[base + 3 * FEAT], vA2 * vA2);
    }
    if (__builtin_expect((unsigned)(gB - g_base) >= 16u, 0)) {
      size_t base = (((size_t)gB * O_DIM + o) * S_MOM) * FEAT + col;
      atomicAdd(&out[base], vB);                  atomicAdd(&out[base + FEAT], vB2);
      atomicAdd(&out[base + 2 * FEAT], vB2 * vB); atomicAdd(&out[base + 3 * FEAT], vB2 * vB2);
    }

    v2f b1, b2, b3, b4;
    b1.x = vA;        b1.y = vB;
    b2.x = vA2;       b2.y = vB2;
    b3.x = vA2 * vA;  b3.y = vB2 * vB;
    b4.x = vA2 * vA2; b4.y = vB2 * vB2;

    acc1 = __builtin_amdgcn_wmma_f32_16x16x4_f32(false, a, false, b1, (short)0, acc1, false, false);
    acc2 = __builtin_amdgcn_wmma_f32_16x16x4_f32(false, a, false, b2, (short)0, acc2, false, false);
    acc3 = __builtin_amdgcn_wmma_f32_16x16x4_f32(false, a, false, b3, (short)0, acc3, false, false);
    acc4 = __builtin_amdgcn_wmma_f32_16x16x4_f32(false, a, false, b4, (short)0, acc4, false, false);
  };

#if HAVE_TDM
  // ---- TDM-staged, double-buffered pipeline ----
  __shared__ float smem[2 * TNODES * FEAT];               // 48 KB
  const int ntiles = (nodes + TNODES - 1) / TNODES;

  if (wave == 0)
    tdm_load_tile(xo + (size_t)c0 * FEAT,
                  (unsigned)(uintptr_t)&smem[0],          // flat addr low 32b = LDS offset
                  n_nodes - c0);

  for (int t = 0; t < ntiles; ++t) {
    const float* fb = &smem[(t & 1) * (TNODES * FEAT)];
    if (wave == 0) __builtin_amdgcn_s_wait_tensorcnt(0);  // tile t landed in LDS
    __syncthreads();
    if (wave == 0 && t + 1 < ntiles) {                    // prefetch tile t+1
      const int ns = c0 + (t + 1) * TNODES;
      tdm_load_tile(xo + (size_t)ns * FEAT,
                    (unsigned)(uintptr_t)&smem[((t + 1) & 1) * (TNODES * FEAT)],
                    n_nodes - ns);
    }
    const int tstart = c0 + t * TNODES;
    const int tend   = min(tstart + TNODES, nfull);
    for (int n0 = tstart; n0 < tend; n0 += 4) {
      const int s = n0 - tstart + 2 * h;                  // local node slots
      step(n0, fabsf(fb[s * FEAT + col]), fabsf(fb[(s + 1) * FEAT + col]));
    }
    __syncthreads();                                      // tile t consumed by all waves
  }
#else
  // ---- fallback: direct per-lane global loads ----
  for (int n0 = c0; n0 < nfull; n0 += 4) {
    const int nA = n0 + 2 * h;
    step(n0, fabsf(xo[(size_t)nA * FEAT + col]),
             fabsf(xo[(size_t)(nA + 1) * FEAT + col]));
  }
#endif
  flush();

  // Scalar tail (never taken for N=100000; half 0 only to avoid duplicate columns)
  if (h == 0) {
    for (int n = nfull; n < c1; ++n) {
      int g = bi[n];
      float v = fabsf(xo[(size_t)n * FEAT + col]);
      float v2 = v * v;
      size_t base = (((size_t)g * O_DIM + o) * S_MOM) * FEAT + col;
      atomicAdd(&out[base], v);
      atomicAdd(&out[base + FEAT], v2);
      atomicAdd(&out[base + 2 * FEAT], v2 * v);
      atomicAdd(&out[base + 3 * FEAT], v2 * v2);
    }
  }
}

extern "C" void kernel_launch(void* const* d_in, const int* in_sizes, int n_in,
                              void* d_out, int out_size, void* d_ws, size_t ws_size,
                              hipStream_t stream) {
  const float* x   = (const float*)d_in[0];
  const int*   bi  = (const int*)d_in[1];   // harness convention: integer -> const int*
  float*       out = (float*)d_out;
  const int n_nodes = in_sizes[1];

  // d_out is poisoned; we accumulate with atomics, so zero it first (stream-ordered).
  zero_kernel<<<(out_size + 511) / 512, 512, 0, stream>>>(out, out_size);

  const int nchunks = (n_nodes + CHUNK - 1) / CHUNK;
  dim3 grid(nchunks, O_DIM);
  moment_seg_wmma<<<grid, 512, 0, stream>>>(x, bi, out, n_nodes);
}